// DiffLogic_82789789597763
// MI455X (gfx1250) — compile-verified
//
#include <hip/hip_runtime.h>

// ---------------------------------------------------------------------------
// DiffLogic network, fully fused for MI455X (gfx1250).
//   Kernel 1 (x3): coef[D,4] = softmax(w[D,16]) @ OP_COEF[16,4]
//                  -> uses V_WMMA_F32_16X16X4_F32 (fp32-exact matrix path)
//   Kernel 2:      whole 3-layer network + group-sum fused per row-tile,
//                  all intermediates live in LDS (h1/h2 rows = 32KB each),
//                  idx/coef metadata stays resident in the 192MB L2.
// ---------------------------------------------------------------------------

typedef __attribute__((ext_vector_type(2))) float v2f;
typedef __attribute__((ext_vector_type(8))) float v8f;

#define D0 1024
#define D1 8192
#define D2 8192
#define D3 10240
#define NCLASS 10
#define TAU 30.0f

#define RROWS 2      // batch rows per workgroup
#define TPB   1024   // threads per block (32 waves)

__constant__ float OPC[16][4] = {
  { 0.f,  0.f,  0.f,  0.f}, { 0.f,  0.f,  0.f,  1.f},
  { 0.f,  1.f,  0.f, -1.f}, { 0.f,  1.f,  0.f,  0.f},
  { 0.f,  0.f,  1.f, -1.f}, { 0.f,  0.f,  1.f,  0.f},
  { 0.f,  1.f,  1.f, -2.f}, { 0.f,  1.f,  1.f, -1.f},
  { 1.f, -1.f, -1.f,  1.f}, { 1.f, -1.f, -1.f,  2.f},
  { 1.f,  0.f, -1.f,  0.f}, { 1.f,  0.f, -1.f,  1.f},
  { 1.f, -1.f,  0.f,  0.f}, { 1.f, -1.f,  0.f,  1.f},
  { 1.f,  0.f,  0.f, -1.f}, { 1.f,  0.f,  0.f,  0.f}
};

// ---------------------------------------------------------------------------
// coef = softmax(w) @ OP_COEF, one 16-neuron tile per wave32.
// A (16x4 f32): lanes 0-15 = M, VGPR0 = K{0 | 2}, VGPR1 = K{1 | 3}
// B (4x16 f32): VGPR0 = rows K{4c+0 | 4c+2}, VGPR1 = rows K{4c+1 | 4c+3}
// D (16x16 f32): VGPR v = row {v | v+8}, N = lane&15; columns 0..3 are coef.
// ---------------------------------------------------------------------------
__global__ __launch_bounds__(256)
void dl_coef_wmma(const float* __restrict__ w, float* __restrict__ coef, int D)
{
    const int wave = (int)((blockIdx.x * blockDim.x + threadIdx.x) >> 5);
    const int lane = (int)(threadIdx.x & 31);
    const int base = wave * 16;                 // neuron tile base
    if (base >= D) return;                      // wave-uniform

    const int m  = lane & 15;                   // matrix row handled by lane
    const int hi = lane >> 4;                   // 0 = lanes 0-15, 1 = 16-31

    // fp32 softmax over the 16 logits of row (base+m); lanes L and L+16
    // redundantly compute the same row (both halves need its probs).
    const float* wr = w + (size_t)(base + m) * 16;
    float p[16];
    float mx = -INFINITY;
#pragma unroll
    for (int k = 0; k < 16; ++k) { p[k] = wr[k]; mx = fmaxf(mx, p[k]); }
    float s = 0.f;
#pragma unroll
    for (int k = 0; k < 16; ++k) { p[k] = expf(p[k] - mx); s += p[k]; }
    const float inv = 1.0f / s;
#pragma unroll
    for (int k = 0; k < 16; ++k) p[k] *= inv;

    v8f d = {0.f, 0.f, 0.f, 0.f, 0.f, 0.f, 0.f, 0.f};
#pragma unroll
    for (int c = 0; c < 4; ++c) {
        const int k0 = 4 * c + 2 * hi;
        v2f A; A.x = p[k0]; A.y = p[k0 + 1];
        v2f B;
        B.x = (m < 4) ? OPC[k0][m]     : 0.f;
        B.y = (m < 4) ? OPC[k0 + 1][m] : 0.f;
        // D = A*B + C  (args: neg_a, A, neg_b, B, c_mod, C, reuse_a, reuse_b)
        d = __builtin_amdgcn_wmma_f32_16x16x4_f32(false, A, false, B,
                                                  (short)0, d, false, false);
    }

    // Columns 0..3 of D are the 4 coefficients of each neuron row.
    if (m < 4) {
#pragma unroll
        for (int v = 0; v < 8; ++v) {
            const int row = v + 8 * hi;
            coef[(size_t)(base + row) * 4 + m] = d[v];
        }
    }
}

// ---------------------------------------------------------------------------
// Fully fused 3-layer network + group-sum. One block = RROWS batch rows.
// LDS: x tile (R*4KB) | h1 tile (R*32KB) | h2 tile (R*32KB) | reduce scratch.
// ---------------------------------------------------------------------------
__global__ __launch_bounds__(TPB, 1)
void dl_fused(const float*  __restrict__ x,
              const int*    __restrict__ ia1, const int* __restrict__ ib1,
              const int*    __restrict__ ia2, const int* __restrict__ ib2,
              const int*    __restrict__ ia3, const int* __restrict__ ib3,
              const float4* __restrict__ c1,
              const float4* __restrict__ c2,
              const float4* __restrict__ c3,
              float* __restrict__ out)
{
    extern __shared__ float smem[];
    float* sh_x   = smem;                        // RROWS * D0
    float* sh_h1  = sh_x  + RROWS * D0;          // RROWS * D1
    float* sh_h2  = sh_h1 + RROWS * D1;          // RROWS * D2
    float* sh_red = sh_h2 + RROWS * D2;          // 32 waves * NCLASS

    const int  tid  = (int)threadIdx.x;
    const long row0 = (long)blockIdx.x * RROWS;

    // ---- Phase A: stage x rows (coalesced; rows are contiguous) ----
    for (int i = tid; i < RROWS * D0; i += TPB)
        sh_x[i] = x[row0 * D0 + i];
    __syncthreads();

    // ---- Phase B: layer 1 (gather from sh_x, write sh_h1) ----
    for (int j = tid; j < D1; j += TPB) {
        const int    ia = ia1[j];
        const int    ib = ib1[j];
        const float4 c  = c1[j];
#pragma unroll
        for (int r = 0; r < RROWS; ++r) {
            const float a = sh_x[r * D0 + ia];
            const float b = sh_x[r * D0 + ib];
            sh_h1[r * D1 + j] = fmaf(c.w, a * b, fmaf(c.z, b, fmaf(c.y, a, c.x)));
        }
    }
    __syncthreads();

    // ---- Phase C: layer 2 (gather from sh_h1, write sh_h2) ----
    for (int j = tid; j < D2; j += TPB) {
        const int    ia = ia2[j];
        const int    ib = ib2[j];
        const float4 c  = c2[j];
#pragma unroll
        for (int r = 0; r < RROWS; ++r) {
            const float a = sh_h1[r * D1 + ia];
            const float b = sh_h1[r * D1 + ib];
            sh_h2[r * D2 + j] = fmaf(c.w, a * b, fmaf(c.z, b, fmaf(c.y, a, c.x)));
        }
    }
    __syncthreads();

    // ---- Phase D: layer 3 fused with group-sum (h3 never materialized) ----
    // 512 threads per row; thread covers j = g*1024 + s*512 + (tid&511).
    const int row = tid >> 9;          // 0..RROWS-1
    const int lr  = tid & 511;
    float acc[NCLASS];
#pragma unroll
    for (int g = 0; g < NCLASS; ++g) acc[g] = 0.f;

    for (int g = 0; g < NCLASS; ++g) {
#pragma unroll
        for (int s = 0; s < 2; ++s) {
            const int    j  = g * 1024 + s * 512 + lr;
            const int    ia = ia3[j];
            const int    ib = ib3[j];
            const float4 c  = c3[j];
            const float  a  = sh_h2[row * D2 + ia];
            const float  b  = sh_h2[row * D2 + ib];
            acc[g] += fmaf(c.w, a * b, fmaf(c.z, b, fmaf(c.y, a, c.x)));
        }
    }

    // wave32 reduction, then cross-wave via LDS
    const int wave = tid >> 5;          // 0..31 (waves 0-15 = row 0)
#pragma unroll
    for (int g = 0; g < NCLASS; ++g) {
        float v = acc[g];
        v += __shfl_xor(v, 16);
        v += __shfl_xor(v, 8);
        v += __shfl_xor(v, 4);
        v += __shfl_xor(v, 2);
        v += __shfl_xor(v, 1);
        if ((tid & 31) == 0) sh_red[wave * NCLASS + g] = v;
    }
    __syncthreads();

    if (tid < RROWS * NCLASS) {
        const int r = tid / NCLASS;
        const int g = tid % NCLASS;
        float s = 0.f;
#pragma unroll
        for (int w = 0; w < 16; ++w)
            s += sh_red[(r * 16 + w) * NCLASS + g];
        out[(row0 + r) * NCLASS + g] = s * (1.0f / TAU);
    }
}

// ---------------------------------------------------------------------------
extern "C" void kernel_launch(void* const* d_in, const int* in_sizes, int n_in,
                              void* d_out, int out_size, void* d_ws, size_t ws_size,
                              hipStream_t stream)
{
    const float* x   = (const float*)d_in[0];
    const float* w1  = (const float*)d_in[1];
    const float* w2  = (const float*)d_in[2];
    const float* w3  = (const float*)d_in[3];
    const int*   ia1 = (const int*)d_in[4];
    const int*   ib1 = (const int*)d_in[5];
    const int*   ia2 = (const int*)d_in[6];
    const int*   ib2 = (const int*)d_in[7];
    const int*   ia3 = (const int*)d_in[8];
    const int*   ib3 = (const int*)d_in[9];
    float*       out = (float*)d_out;

    float* ws = (float*)d_ws;
    float* c1 = ws;                    // D1*4 floats
    float* c2 = c1 + (size_t)D1 * 4;   // D2*4 floats
    float* c3 = c2 + (size_t)D2 * 4;   // D3*4 floats

    // coefficient precompute via WMMA: one wave per 16 neurons, 8 waves/block
    dl_coef_wmma<<<D1 / 128, 256, 0, stream>>>(w1, c1, D1);
    dl_coef_wmma<<<D2 / 128, 256, 0, stream>>>(w2, c2, D2);
    dl_coef_wmma<<<D3 / 128, 256, 0, stream>>>(w3, c3, D3);

    // fused network: 4096 / RROWS blocks, dynamic LDS for x|h1|h2|reduce
    const size_t shmem =
        (size_t)(RROWS * D0 + RROWS * D1 + RROWS * D2 + 32 * NCLASS) * sizeof(float);
    dl_fused<<<4096 / RROWS, TPB, shmem, stream>>>(
        x, ia1, ib1, ia2, ib2, ia3, ib3,
        (const float4*)c1, (const float4*)c2, (const float4*)c3, out);
}